// AffinityGAT_41068477284360
// MI455X (gfx1250) — compile-verified
//
#include <hip/hip_runtime.h>
#include <hip/hip_bf16.h>
#include <math.h>

typedef __attribute__((ext_vector_type(2))) float v2f;
typedef __attribute__((ext_vector_type(8))) float v8f;

#define NEG_SLOPE 0.2f
#define LDA_STRIDE 132   // 128+4: A rows land on distinct banks (132 % 64 == 4)
#define LDB_STRIDE 136   // 128+8: B row pairs use disjoint bank ranges (136 % 64 == 8)

// ---------------- float atomic max via int bit tricks (always compiles) ----
__device__ __forceinline__ void atomicMaxF32(float* addr, float value) {
  if (value >= 0.0f) {
    atomicMax((int*)addr, __float_as_int(value));
  } else {
    atomicMin((unsigned int*)addr, __float_as_uint(value));
  }
}

// ---------------- WMMA f32 GEMM: C[N,128] = A[N,128] * B[128,128] ----------
// block = 256 threads = 8 waves; block handles one 16-row tile of A;
// wave w computes the 16x16 output tile at columns w*16.
// A tile (16x128) and full B (128x128) staged in LDS with padded strides.
__global__ __launch_bounds__(256) void gemm128_wmma(
    const float* __restrict__ A, const float* __restrict__ B,
    float* __restrict__ C, int N) {
  __shared__ float lA[16 * LDA_STRIDE];
  __shared__ float lB[128 * LDB_STRIDE];

  const int t = threadIdx.x;
  const int lane = t & 31;
  const int w = t >> 5;                      // wave in block == column tile
  const int m0 = blockIdx.x * 16;
  const int n0 = w * 16;
  const int hi  = lane >> 4;                 // 0: K {0,1}, 1: K {2,3}
  const int l15 = lane & 15;

  // ---- cooperative stage: A tile, 512 float4s (2 per thread), coalesced
#pragma unroll
  for (int i = 0; i < 2; ++i) {
    const int idx = t + i * 256;             // float4 index
    const int row = idx >> 5;                // 32 float4 per row
    const int c4 = idx & 31;
    int gr = m0 + row; if (gr >= N) gr = N - 1;
    const float4 v = *(const float4*)(A + (long)gr * 128 + c4 * 4);
    *(float4*)(lA + row * LDA_STRIDE + c4 * 4) = v;
  }
  // ---- cooperative stage: B 128x128, 4096 float4s (16 per thread), coalesced
#pragma unroll
  for (int i = 0; i < 16; ++i) {
    const int idx = t + i * 256;
    const int row = idx >> 5;
    const int c4 = idx & 31;
    const float4 v = *(const float4*)(B + row * 128 + c4 * 4);
    *(float4*)(lB + row * LDB_STRIDE + c4 * 4) = v;
  }
  __syncthreads();

  v8f acc = {};
  for (int k = 0; k < 128; k += 4) {
    const int ka = k + hi * 2;
    v2f a, b;
    a.x = lA[l15 * LDA_STRIDE + ka];
    a.y = lA[l15 * LDA_STRIDE + ka + 1];
    b.x = lB[ka * LDB_STRIDE + n0 + l15];
    b.y = lB[(ka + 1) * LDB_STRIDE + n0 + l15];
    acc = __builtin_amdgcn_wmma_f32_16x16x4_f32(
        /*neg_a=*/false, a, /*neg_b=*/false, b,
        /*c_mod=*/(short)0, acc, /*reuse_a=*/false, /*reuse_b=*/false);
  }

  if (m0 + 16 <= N) {                        // fast path: full tile, no guards
#pragma unroll
    for (int j = 0; j < 8; ++j)
      C[(long)(m0 + j + hi * 8) * 128 + n0 + l15] = acc[j];
  } else {
#pragma unroll
    for (int j = 0; j < 8; ++j) {
      const int row = m0 + j + hi * 8;
      if (row < N) C[(long)row * 128 + n0 + l15] = acc[j];
    }
  }
}

// ---------------- per-node attention scalars: as = h.a_s, ad = h.a_d -------
// one wave per node, float4 per lane (coalesced 512B row)
__global__ __launch_bounds__(256) void node_alphas128(
    const float* __restrict__ h, const float* __restrict__ a_s,
    const float* __restrict__ a_d, float* __restrict__ as_out,
    float* __restrict__ ad_out, int N) {
  const int lane = threadIdx.x & 31;
  const int n = (blockIdx.x * blockDim.x + threadIdx.x) >> 5;
  if (n >= N) return;
  const float4 hv = ((const float4*)(h + (long)n * 128))[lane];
  const float4 sv = ((const float4*)a_s)[lane];
  const float4 dv = ((const float4*)a_d)[lane];
  float ps = hv.x * sv.x + hv.y * sv.y + hv.z * sv.z + hv.w * sv.w;
  float pd = hv.x * dv.x + hv.y * dv.y + hv.z * dv.z + hv.w * dv.w;
#pragma unroll
  for (int off = 16; off > 0; off >>= 1) {
    ps += __shfl_xor(ps, off, 32);
    pd += __shfl_xor(pd, off, 32);
  }
  if (lane == 0) { as_out[n] = ps; ad_out[n] = pd; }
}

// ---------------- layer 2: h1 = h @ W[:,0]; as = h1*a_s0; ad = h1*a_d0 -----
__global__ __launch_bounds__(256) void node_proj1(
    const float* __restrict__ h, const float* __restrict__ W,
    const float* __restrict__ a_s, const float* __restrict__ a_d,
    float* __restrict__ hl, float* __restrict__ as_out,
    float* __restrict__ ad_out, int N) {
  const int lane = threadIdx.x & 31;
  const int n = (blockIdx.x * blockDim.x + threadIdx.x) >> 5;
  if (n >= N) return;
  const float4 hv = ((const float4*)(h + (long)n * 128))[lane];
  const float4 wv = ((const float4*)W)[lane];
  float p = hv.x * wv.x + hv.y * wv.y + hv.z * wv.z + hv.w * wv.w;
#pragma unroll
  for (int off = 16; off > 0; off >>= 1) p += __shfl_xor(p, off, 32);
  if (lane == 0) {
    hl[n] = p;
    as_out[n] = p * a_s[0];
    ad_out[n] = p * a_d[0];
  }
}

// ---------------- init kernels ---------------------------------------------
__global__ void init_node_scalars(float* __restrict__ m, float* __restrict__ dn, int N) {
  const int i = blockIdx.x * blockDim.x + threadIdx.x;
  if (i < N) { m[i] = -INFINITY; dn[i] = 0.0f; }
}
__global__ void init_out_bias(float* __restrict__ out, const float* __restrict__ b,
                              int N, int C) {
  const long i = (long)blockIdx.x * blockDim.x + threadIdx.x;
  if (i < (long)N * C) out[i] = b[i % C];
}
__global__ void init_zero(float* __restrict__ p, int n) {
  const int i = blockIdx.x * blockDim.x + threadIdx.x;
  if (i < n) p[i] = 0.0f;
}

// ---------------- edge pass 1: leaky-relu logits + segment max -------------
__global__ void edge_logits(const int* __restrict__ ei,
                            const float* __restrict__ as,
                            const float* __restrict__ ad,
                            float* __restrict__ ebuf, float* __restrict__ m,
                            int E, int N) {
  const int e = blockIdx.x * blockDim.x + threadIdx.x;
  const int ET = E + N;
  if (e >= ET) return;
  int s, d;
  if (e < E) { s = ei[e]; d = ei[E + e]; } else { s = e - E; d = s; }
  float v = as[s] + ad[d];
  v = (v > 0.0f) ? v : NEG_SLOPE * v;
  ebuf[e] = v;
  atomicMaxF32(&m[d], v);
}

// ---------------- edge pass 2: exp + segment sum ---------------------------
__global__ void edge_expsum(const int* __restrict__ ei,
                            const float* __restrict__ m,
                            float* __restrict__ ebuf, float* __restrict__ dn,
                            int E, int N) {
  const int e = blockIdx.x * blockDim.x + threadIdx.x;
  const int ET = E + N;
  if (e >= ET) return;
  int d;
  if (e < E) d = ei[E + e]; else d = e - E;
  const float w = expf(ebuf[e] - m[d]);
  ebuf[e] = w;
  atomicAdd(&dn[d], w);
}

// ---------------- edge pass 3: weighted aggregation (C=128) ----------------
// one wave per edge; lane handles 4 contiguous channels (b128 gather)
__global__ __launch_bounds__(256) void edge_agg128(
    const int* __restrict__ ei, const float* __restrict__ ebuf,
    const float* __restrict__ dn, const float* __restrict__ hl,
    float* __restrict__ out, int E, int N) {
  const int lane = threadIdx.x & 31;
  const int e = (blockIdx.x * blockDim.x + threadIdx.x) >> 5;
  const int ET = E + N;
  if (e >= ET) return;
  int s, d;
  if (e < E) { s = ei[e]; d = ei[E + e]; } else { s = e - E; d = s; }
  const float coef = ebuf[e] / dn[d];
  const float4 hv = ((const float4*)(hl + (long)s * 128))[lane];
  float* o = out + (long)d * 128 + lane * 4;
  atomicAdd(o + 0, coef * hv.x);
  atomicAdd(o + 1, coef * hv.y);
  atomicAdd(o + 2, coef * hv.z);
  atomicAdd(o + 3, coef * hv.w);
}

// ---------------- edge aggregation (C=1) -----------------------------------
__global__ void edge_agg1(const int* __restrict__ ei, const float* __restrict__ ebuf,
                          const float* __restrict__ dn, const float* __restrict__ hl,
                          float* __restrict__ out, int E, int N) {
  const int e = blockIdx.x * blockDim.x + threadIdx.x;
  const int ET = E + N;
  if (e >= ET) return;
  int s, d;
  if (e < E) { s = ei[e]; d = ei[E + e]; } else { s = e - E; d = s; }
  atomicAdd(&out[d], (ebuf[e] / dn[d]) * hl[s]);
}

// ---------------- relu in place --------------------------------------------
__global__ void relu_inplace(float* __restrict__ x, long n) {
  const long i = (long)blockIdx.x * blockDim.x + threadIdx.x;
  if (i < n) x[i] = fmaxf(x[i], 0.0f);
}

// ---------------- global mean pool -----------------------------------------
__global__ void node_pool(const float* __restrict__ h2, const int* __restrict__ batch,
                          float* __restrict__ psum, float* __restrict__ pcnt, int N) {
  const int i = blockIdx.x * blockDim.x + threadIdx.x;
  if (i >= N) return;
  const int g = batch[i];
  atomicAdd(&psum[g], h2[i]);
  atomicAdd(&pcnt[g], 1.0f);
}
__global__ void pool_final(const float* __restrict__ psum, const float* __restrict__ pcnt,
                           float* __restrict__ out, int G) {
  const int g = blockIdx.x * blockDim.x + threadIdx.x;
  if (g < G) out[g] = psum[g] / fmaxf(pcnt[g], 1.0f);
}

// ===========================================================================
extern "C" void kernel_launch(void* const* d_in, const int* in_sizes, int n_in,
                              void* d_out, int out_size, void* d_ws, size_t ws_size,
                              hipStream_t stream) {
  const float* x   = (const float*)d_in[0];
  const int* ei    = (const int*)d_in[1];
  const int* batch = (const int*)d_in[2];
  const float* W0  = (const float*)d_in[3];
  const float* W1  = (const float*)d_in[4];
  const float* W2  = (const float*)d_in[5];
  const float* as0 = (const float*)d_in[6];
  const float* as1 = (const float*)d_in[7];
  const float* as2 = (const float*)d_in[8];
  const float* ad0 = (const float*)d_in[9];
  const float* ad1 = (const float*)d_in[10];
  const float* ad2 = (const float*)d_in[11];
  const float* b0  = (const float*)d_in[12];
  const float* b1  = (const float*)d_in[13];
  const float* b2  = (const float*)d_in[14];

  const int N = in_sizes[0] / 128;
  const int E = in_sizes[1] / 2;
  const int G = out_size;
  const int ET = E + N;

  // workspace carve (floats)
  float* ws   = (float*)d_ws;
  float* bufA = ws;                         // N*128  (h_lin)
  float* bufB = bufA + (long)N * 128;       // N*128  (aggregated out / next h)
  float* asv  = bufB + (long)N * 128;       // N
  float* adv  = asv + N;                    // N
  float* mbuf = adv + N;                    // N
  float* dbuf = mbuf + N;                   // N
  float* ebuf = dbuf + N;                   // E+N
  float* h2   = ebuf + ET;                  // N
  float* out2 = h2 + N;                     // N
  float* psum = out2 + N;                   // G
  float* pcnt = psum + G;                   // G

  const int T = 256;
  const int gemm_blocks = (N + 15) / 16;    // one 16-row tile per block
  const int waveN_blocks = (N * 32 + T - 1) / T;
  const int nodeBlocks = (N + T - 1) / T;
  const int edgeBlocks = (ET + T - 1) / T;
  const int edgeWaveBlocks = ((long)ET * 32 + T - 1) / T;
  const long NC = (long)N * 128;
  const int ncBlocks = (int)((NC + T - 1) / T);

  // ---- Layer 0: x -> bufA -> (attn) -> bufB, relu
  gemm128_wmma<<<gemm_blocks, T, 0, stream>>>(x, W0, bufA, N);
  node_alphas128<<<waveN_blocks, T, 0, stream>>>(bufA, as0, ad0, asv, adv, N);
  init_node_scalars<<<nodeBlocks, T, 0, stream>>>(mbuf, dbuf, N);
  init_out_bias<<<ncBlocks, T, 0, stream>>>(bufB, b0, N, 128);
  edge_logits<<<edgeBlocks, T, 0, stream>>>(ei, asv, adv, ebuf, mbuf, E, N);
  edge_expsum<<<edgeBlocks, T, 0, stream>>>(ei, mbuf, ebuf, dbuf, E, N);
  edge_agg128<<<edgeWaveBlocks, T, 0, stream>>>(ei, ebuf, dbuf, bufA, bufB, E, N);
  relu_inplace<<<ncBlocks, T, 0, stream>>>(bufB, NC);

  // ---- Layer 1: bufB -> bufA -> (attn) -> bufB, relu
  gemm128_wmma<<<gemm_blocks, T, 0, stream>>>(bufB, W1, bufA, N);
  node_alphas128<<<waveN_blocks, T, 0, stream>>>(bufA, as1, ad1, asv, adv, N);
  init_node_scalars<<<nodeBlocks, T, 0, stream>>>(mbuf, dbuf, N);
  init_out_bias<<<ncBlocks, T, 0, stream>>>(bufB, b1, N, 128);
  edge_logits<<<edgeBlocks, T, 0, stream>>>(ei, asv, adv, ebuf, mbuf, E, N);
  edge_expsum<<<edgeBlocks, T, 0, stream>>>(ei, mbuf, ebuf, dbuf, E, N);
  edge_agg128<<<edgeWaveBlocks, T, 0, stream>>>(ei, ebuf, dbuf, bufA, bufB, E, N);
  relu_inplace<<<ncBlocks, T, 0, stream>>>(bufB, NC);

  // ---- Layer 2 (C_out = 1): bufB -> h2 -> out2
  node_proj1<<<waveN_blocks, T, 0, stream>>>(bufB, W2, as2, ad2, h2, asv, adv, N);
  init_node_scalars<<<nodeBlocks, T, 0, stream>>>(mbuf, dbuf, N);
  init_out_bias<<<nodeBlocks, T, 0, stream>>>(out2, b2, N, 1);
  edge_logits<<<edgeBlocks, T, 0, stream>>>(ei, asv, adv, ebuf, mbuf, E, N);
  edge_expsum<<<edgeBlocks, T, 0, stream>>>(ei, mbuf, ebuf, dbuf, E, N);
  edge_agg1<<<edgeBlocks, T, 0, stream>>>(ei, ebuf, dbuf, h2, out2, E, N);

  // ---- Global mean pool
  init_zero<<<(2 * G + T - 1) / T, T, 0, stream>>>(psum, 2 * G);
  node_pool<<<nodeBlocks, T, 0, stream>>>(out2, batch, psum, pcnt, N);
  pool_final<<<(G + T - 1) / T, T, 0, stream>>>(psum, pcnt, (float*)d_out, G);

  (void)n_in; (void)ws_size;
}